// RepmetLoss_33097017983474
// MI455X (gfx1250) — compile-verified
//
#include <hip/hip_runtime.h>

typedef float v2f __attribute__((ext_vector_type(2)));
typedef float v8f __attribute__((ext_vector_type(8)));

#define EMBD  256
#define NREPS 2000
#define NCLS  400
#define NSAMP 32768
#define TILES 125            // 2000 / 16 column tiles, exact
#define XSTR  260            // padded LDS row stride (260 % 64 == 4 -> conflict-free frag loads)
#define ROWS  16
#define NBLK  (NSAMP / ROWS) // 2048 workgroups

// Pass 0: per-rep squared norms (2000 values) into workspace.
__global__ __launch_bounds__(256) void repmet_rnorm(const float* __restrict__ reps,
                                                    float* __restrict__ rn) {
  int j = blockIdx.x * 256 + threadIdx.x;
  if (j < NREPS) {
    const float4* p = (const float4*)(reps + (size_t)j * EMBD);
    float s = 0.f;
#pragma unroll 8
    for (int i = 0; i < EMBD / 4; ++i) {
      float4 v = p[i];
      s += v.x * v.x + v.y * v.y + v.z * v.z + v.w * v.w;
    }
    rn[j] = s;
  }
}

// Pass 1: fused distances (fp32 WMMA) + per-sample min_cor/min_inc + class prob sums + argmax.
__global__ __launch_bounds__(256) void repmet_main(
    const float* __restrict__ x, const int* __restrict__ target,
    const float* __restrict__ reps, const float* __restrict__ rnorm2,
    float* __restrict__ losses_out, float* __restrict__ pred_out,
    float* __restrict__ loss_part, float* __restrict__ acc_part) {
  __shared__ float xs[ROWS * XSTR];        // 16.6 KB x tile
  __shared__ float numacc[ROWS * NCLS];    // 25.6 KB per-class kernel sums
  __shared__ int   minci[ROWS];            // own-class min dist (bits; dist>=0 -> int order ok)
  __shared__ int   minii[ROWS];            // other-class min dist
  __shared__ float xn2[ROWS];
  __shared__ int   tgts[ROWS];
  __shared__ float fl[ROWS], fa[ROWS];

  const int tid  = threadIdx.x;
  const int row0 = blockIdx.x * ROWS;

  // Stage x tile into LDS (coalesced float4).
  for (int i = tid; i < ROWS * (EMBD / 4); i += 256) {
    int r = i >> 6;
    int c = i & 63;
    float4 v = ((const float4*)(x + (size_t)(row0 + r) * EMBD))[c];
    float* d = &xs[r * XSTR + c * 4];
    d[0] = v.x; d[1] = v.y; d[2] = v.z; d[3] = v.w;
  }
  for (int i = tid; i < ROWS * NCLS; i += 256) numacc[i] = 0.f;
  if (tid < ROWS) {
    minci[tid] = __float_as_int(3.0e38f);
    minii[tid] = __float_as_int(3.0e38f);
    tgts[tid]  = target[row0 + tid];
  }
  __syncthreads();
  if (tid < ROWS) {   // per-sample squared norm from the staged tile
    float s = 0.f;
    for (int k = 0; k < EMBD; ++k) { float v = xs[tid * XSTR + k]; s += v * v; }
    xn2[tid] = s;
  }
  __syncthreads();

  const int wave = tid >> 5;   // 8 wave32 waves
  const int lane = tid & 31;
  const int lrow = lane & 15;  // fragment row (M for A, N for B)
  const int lhi  = lane >> 4;  // which K half-pair

  const float* arow = &xs[lrow * XSTR + lhi * 2];

  for (int t = wave; t < TILES; t += 8) {
    const int jbase = t * 16;
    const int j     = jbase + lrow;
    const float* brow = reps + (size_t)j * EMBD + lhi * 2;
    if (t + 8 < TILES)  // CDNA5 global_prefetch_b8 for next B tile
      __builtin_prefetch(reps + (size_t)(j + 128) * EMBD, 0, 1);

    v8f acc = {};
#pragma unroll 8
    for (int k = 0; k < EMBD; k += 4) {
      v2f a = *(const v2f*)(arow + k);   // ds_load_b64, bank-conflict-free
      v2f b = *(const v2f*)(brow + k);   // global_load_b64 from L2-resident reps
      acc = __builtin_amdgcn_wmma_f32_16x16x4_f32(false, a, false, b,
                                                  (short)0, acc, false, false);
    }

    const float rn  = rnorm2[j];
    const int   cls = j / 5;
#pragma unroll
    for (int v = 0; v < 8; ++v) {
      const int   m    = v + 8 * lhi;                 // sample row in tile
      float       d2   = xn2[m] + rn - 2.0f * acc[v]; // squared distance
      d2 = fmaxf(d2, 0.0f);
      const float dist = sqrtf(d2);
      const float prob = __expf(-2.0f * d2);          // exp(-d^2/(2*0.5^2))
      if (cls == tgts[m]) atomicMin(&minci[m], __float_as_int(dist));
      else                atomicMin(&minii[m], __float_as_int(dist));
      atomicAdd(&numacc[m * NCLS + cls], prob);       // ds_add_f32
    }
  }
  __syncthreads();

  if (tid < ROWS) {
    const int s = tid;
    float best = -1.0f; int bidx = 0;
    for (int c = 0; c < NCLS; ++c) {      // first-max semantics like jnp.argmax
      float vv = numacc[s * NCLS + c];
      if (vv > best) { best = vv; bidx = c; }
    }
    const float mc   = __int_as_float(minci[s]);
    const float mi   = __int_as_float(minii[s]);
    const float loss = fmaxf(mc - mi + 1.0f, 0.0f);
    losses_out[row0 + s] = loss;
    pred_out[row0 + s]   = (float)bidx;
    fl[s] = loss;
    fa[s] = (bidx == tgts[s]) ? 1.0f : 0.0f;
  }
  __syncthreads();
  if (tid == 0) {
    float sl = 0.f, sa = 0.f;
    for (int i = 0; i < ROWS; ++i) { sl += fl[i]; sa += fa[i]; }
    loss_part[blockIdx.x] = sl;
    acc_part[blockIdx.x]  = sa;
  }
}

// Pass 2: deterministic reduction of per-block partials -> total_loss, acc.
__global__ __launch_bounds__(256) void repmet_final(const float* __restrict__ lp,
                                                    const float* __restrict__ ap,
                                                    float* __restrict__ out) {
  __shared__ float sl[256], sa[256];
  float a = 0.f, b = 0.f;
  for (int i = threadIdx.x; i < NBLK; i += 256) { a += lp[i]; b += ap[i]; }
  sl[threadIdx.x] = a; sa[threadIdx.x] = b;
  __syncthreads();
  for (int s = 128; s > 0; s >>= 1) {
    if ((int)threadIdx.x < s) {
      sl[threadIdx.x] += sl[threadIdx.x + s];
      sa[threadIdx.x] += sa[threadIdx.x + s];
    }
    __syncthreads();
  }
  if (threadIdx.x == 0) {
    out[0]             = sl[0] / (float)NSAMP;  // total_loss
    out[1 + 2 * NSAMP] = sa[0] / (float)NSAMP;  // acc at index 65537
  }
}

extern "C" void kernel_launch(void* const* d_in, const int* in_sizes, int n_in,
                              void* d_out, int out_size, void* d_ws, size_t ws_size,
                              hipStream_t stream) {
  const float* x      = (const float*)d_in[0];
  const int*   target = (const int*)d_in[1];
  const float* reps   = (const float*)d_in[2];
  float* out = (float*)d_out;
  float* ws  = (float*)d_ws;
  float* rn  = ws;          // 2000 (pad 2048) rep norms
  float* lp  = ws + 2048;   // 2048 loss partials
  float* ap  = ws + 4096;   // 2048 acc partials

  repmet_rnorm<<<(NREPS + 255) / 256, 256, 0, stream>>>(reps, rn);
  repmet_main<<<NBLK, 256, 0, stream>>>(x, target, reps, rn,
                                        out + 1, out + 1 + NSAMP, lp, ap);
  repmet_final<<<1, 256, 0, stream>>>(lp, ap, out);
}